// EncodingModule2d_51290499448905
// MI455X (gfx1250) — compile-verified
//
#include <hip/hip_runtime.h>
#include <hip/hip_bf16.h>
#include <math.h>

#define D_CH   256
#define NPIX   4096   // 64*64
#define KCENT  32
#define BATCH  8
#define EPSV   1e-5f

typedef __attribute__((ext_vector_type(2))) float v2f;
typedef __attribute__((ext_vector_type(8))) float v8f;

// ---------------------------------------------------------------------------
// WMMA wrapper: D = A(16x4 f32) * B(4x16 f32) + C(16x16 f32)
// A layout (ISA 7.12.2): lane L (r=L&15, half=L>>4) holds row M=r,
//   vgpr0 = A[r][2*half], vgpr1 = A[r][2*half+1]
// B layout: vgpr0 = B[2*half][r], vgpr1 = B[2*half+1][r]
// C layout: vgpr j = C[j + 8*half][r]
// ---------------------------------------------------------------------------
__device__ __forceinline__ v8f wmma_f32(v2f a, v2f b, v8f c) {
    return __builtin_amdgcn_wmma_f32_16x16x4_f32(false, a, false, b,
                                                 (short)0, c, false, false);
}

// ---------------------------------------------------------------------------
// K0: zero the z accumulator (B*D floats)
// ---------------------------------------------------------------------------
__global__ void init_zero(float* __restrict__ p, int n) {
    int i = blockIdx.x * blockDim.x + threadIdx.x;
    if (i < n) p[i] = 0.0f;
}

// ---------------------------------------------------------------------------
// K1: stem 1x1 conv GEMM + BN2 + ReLU, LDS-blocked.
// Workgroup (128 thr / 4 waves) computes a 64(d) x 64(n) tile of
//   S[b][o][n] = relu(BN2(sum_c conv_w[o][c] * x[b][c][n]))
// Each wave: 32x32 register block = 2x2 WMMA C-tiles.
// LDS: As[d][k] 64x16 (pad 18), Bs[n][k] 64x16 transposed (pad 18).
// grid (NPIX/64, D/64, B), block 128
// ---------------------------------------------------------------------------
#define TM  64
#define TN  64
#define TK  16
#define LDP 18   // padded row stride (floats): even (8B-aligned v2f), 18*r mod 64 distinct for r<16

__global__ __launch_bounds__(128) void stem_gemm(
        const float* __restrict__ x,
        const float* __restrict__ conv_w,
        const float* __restrict__ bn2_g,
        const float* __restrict__ bn2_b,
        const float* __restrict__ bn2_m,
        const float* __restrict__ bn2_v,
        float* __restrict__ s) {
    __shared__ float As[TM * LDP];   // As[dLocal][kLocal]
    __shared__ float Bs[TN * LDP];   // Bs[nLocal][kLocal] (transposed chunk of x)

    const int tid  = threadIdx.x;    // 0..127
    const int lane = tid & 31;
    const int wave = tid >> 5;       // 0..3
    const int wr   = wave >> 1;      // wave row (d direction)
    const int wc   = wave & 1;       // wave col (n direction)
    const int half = lane >> 4;
    const int r    = lane & 15;

    const int n0 = blockIdx.x * TN;
    const int d0 = blockIdx.y * TM;
    const int b  = blockIdx.z;
    const float* __restrict__ xb = x + (size_t)b * D_CH * NPIX;

    // per-thread global staging assignments
    const int a_row = tid >> 1;          // 0..63  (d local)
    const int a_col = (tid & 1) * 8;     // 0 or 8 (k local)
    const int b_row = tid >> 3;          // 0..15  (k local)
    const int b_col = (tid & 7) * 8;     // 0..56  (n local)

    v8f acc[2][2] = {};

    // prefetch chunk 0 into registers
    float4 av0 = *(const float4*)(conv_w + (size_t)(d0 + a_row) * D_CH + a_col);
    float4 av1 = *(const float4*)(conv_w + (size_t)(d0 + a_row) * D_CH + a_col + 4);
    float4 bv0 = *(const float4*)(xb + (size_t)(b_row) * NPIX + n0 + b_col);
    float4 bv1 = *(const float4*)(xb + (size_t)(b_row) * NPIX + n0 + b_col + 4);

    for (int kk = 0; kk < D_CH; kk += TK) {
        __syncthreads();   // previous chunk fully consumed
        *(float4*)(&As[a_row * LDP + a_col])     = av0;
        *(float4*)(&As[a_row * LDP + a_col + 4]) = av1;
        Bs[(b_col + 0) * LDP + b_row] = bv0.x;
        Bs[(b_col + 1) * LDP + b_row] = bv0.y;
        Bs[(b_col + 2) * LDP + b_row] = bv0.z;
        Bs[(b_col + 3) * LDP + b_row] = bv0.w;
        Bs[(b_col + 4) * LDP + b_row] = bv1.x;
        Bs[(b_col + 5) * LDP + b_row] = bv1.y;
        Bs[(b_col + 6) * LDP + b_row] = bv1.z;
        Bs[(b_col + 7) * LDP + b_row] = bv1.w;
        __syncthreads();

        // prefetch next chunk while computing this one (uniform branch)
        const int kn = kk + TK;
        if (kn < D_CH) {
            av0 = *(const float4*)(conv_w + (size_t)(d0 + a_row) * D_CH + kn + a_col);
            av1 = *(const float4*)(conv_w + (size_t)(d0 + a_row) * D_CH + kn + a_col + 4);
            bv0 = *(const float4*)(xb + (size_t)(kn + b_row) * NPIX + n0 + b_col);
            bv1 = *(const float4*)(xb + (size_t)(kn + b_row) * NPIX + n0 + b_col + 4);
        }

#pragma unroll
        for (int ks = 0; ks < TK; ks += 4) {
            v2f af0 = *(const v2f*)(&As[(wr * 32 +  0 + r) * LDP + ks + 2 * half]);
            v2f af1 = *(const v2f*)(&As[(wr * 32 + 16 + r) * LDP + ks + 2 * half]);
            v2f bf0 = *(const v2f*)(&Bs[(wc * 32 +  0 + r) * LDP + ks + 2 * half]);
            v2f bf1 = *(const v2f*)(&Bs[(wc * 32 + 16 + r) * LDP + ks + 2 * half]);
            acc[0][0] = wmma_f32(af0, bf0, acc[0][0]);
            acc[0][1] = wmma_f32(af0, bf1, acc[0][1]);
            acc[1][0] = wmma_f32(af1, bf0, acc[1][0]);
            acc[1][1] = wmma_f32(af1, bf1, acc[1][1]);
        }
    }

    // epilogue: BN2 + ReLU + store
    float* __restrict__ sb = s + (size_t)b * D_CH * NPIX;
#pragma unroll
    for (int mi = 0; mi < 2; ++mi) {
#pragma unroll
        for (int j = 0; j < 8; ++j) {
            const int d = d0 + wr * 32 + mi * 16 + j + 8 * half;
            const float sc = bn2_g[d] * rsqrtf(bn2_v[d] + EPSV);
            const float mn = bn2_m[d];
            const float bb = bn2_b[d];
#pragma unroll
            for (int ni = 0; ni < 2; ++ni) {
                const int n = n0 + wc * 32 + ni * 16 + r;
                const float v = (acc[mi][ni][j] - mn) * sc + bb;
                sb[(size_t)d * NPIX + n] = fmaxf(v, 0.0f);
            }
        }
    }
}

// ---------------------------------------------------------------------------
// K2: yc = centers @ S_b (both 16-row k-tiles in one wave), y2 from the same
// B-operand loads, softmax over K=32, store a[b][k][n] (K-major).
// grid (NPIX/16, B), block 32
// ---------------------------------------------------------------------------
__global__ void assign_softmax(const float* __restrict__ s,
                               const float* __restrict__ centers,
                               const float* __restrict__ scales,
                               float* __restrict__ a_out) {
    const int lane = threadIdx.x & 31;
    const int half = lane >> 4;
    const int r    = lane & 15;
    const int n0   = blockIdx.x * 16;
    const int b    = blockIdx.y;

    const float* __restrict__ sb = s + (size_t)b * D_CH * NPIX;
    v8f acc0 = {};   // centers rows 0..15
    v8f acc1 = {};   // centers rows 16..31
    float y2p = 0.0f;
    for (int dd = 0; dd < D_CH; dd += 4) {
        v2f a0 = *(const v2f*)(centers + (size_t)(r)      * D_CH + dd + 2 * half);
        v2f a1 = *(const v2f*)(centers + (size_t)(16 + r) * D_CH + dd + 2 * half);
        v2f bm;
        bm.x = sb[(size_t)(dd + 2 * half + 0) * NPIX + n0 + r];
        bm.y = sb[(size_t)(dd + 2 * half + 1) * NPIX + n0 + r];
        acc0 = wmma_f32(a0, bm, acc0);
        acc1 = wmma_f32(a1, bm, acc1);
        y2p += bm.x * bm.x + bm.y * bm.y;
    }
    y2p += __shfl_xor(y2p, 16, 32);           // full ||y||^2 for column n0+r

    // ||c_k||^2: lane computes its own row k = lane, broadcast on demand
    float c2own = 0.0f;
    for (int d = 0; d < D_CH; ++d) {
        const float cv = centers[(size_t)lane * D_CH + d];
        c2own += cv * cv;
    }

    // logits for this lane: k = j + 8*half + 16*t  (16 of the 32 per lane)
    float lg[16];
    float mx = -INFINITY;
#pragma unroll
    for (int t = 0; t < 2; ++t) {
#pragma unroll
        for (int j = 0; j < 8; ++j) {
            const int k  = j + 8 * half + 16 * t;
            const float yc  = (t == 0) ? acc0[j] : acc1[j];
            const float c2k = __shfl(c2own, k, 32);
            const float v   = scales[k] * (y2p - 2.0f * yc + c2k);
            lg[j + 8 * t] = v;
            mx = fmaxf(mx, v);
        }
    }
    mx = fmaxf(mx, __shfl_xor(mx, 16, 32));
    float sum = 0.0f;
#pragma unroll
    for (int i = 0; i < 16; ++i) { lg[i] = __expf(lg[i] - mx); sum += lg[i]; }
    sum += __shfl_xor(sum, 16, 32);
    const float inv = 1.0f / sum;

    float* __restrict__ ab = a_out + (size_t)b * KCENT * NPIX;
#pragma unroll
    for (int t = 0; t < 2; ++t) {
#pragma unroll
        for (int j = 0; j < 8; ++j) {
            const int k = j + 8 * half + 16 * t;
            ab[(size_t)k * NPIX + n0 + r] = lg[j + 8 * t] * inv;
        }
    }
}

// ---------------------------------------------------------------------------
// K3: aggY = a(32x4096) @ y(4096x256), asum[k] from the same A loads,
// agg = aggY - asum*centers, BN1 + ReLU, atomic K-sum into z[b][d].
// grid (D/16, KCENT/16, B), block 32
// ---------------------------------------------------------------------------
__global__ void agg_bn_pool(const float* __restrict__ a_mat,
                            const float* __restrict__ s,
                            const float* __restrict__ centers,
                            const float* __restrict__ bn1_g,
                            const float* __restrict__ bn1_b,
                            const float* __restrict__ bn1_m,
                            const float* __restrict__ bn1_v,
                            float* __restrict__ z) {
    const int lane = threadIdx.x & 31;
    const int half = lane >> 4;
    const int r    = lane & 15;
    const int d0   = blockIdx.x * 16;
    const int k0   = blockIdx.y * 16;
    const int b    = blockIdx.z;

    const float* __restrict__ ab = a_mat + (size_t)b * KCENT * NPIX;
    const float* __restrict__ sb = s     + (size_t)b * D_CH  * NPIX;
    v8f acc = {};
    float asump = 0.0f;
    for (int nn = 0; nn < NPIX; nn += 4) {
        v2f a  = *(const v2f*)(ab + (size_t)(k0 + r) * NPIX + nn + 2 * half);
        v2f bm = *(const v2f*)(sb + (size_t)(d0 + r) * NPIX + nn + 2 * half);
        asump += a.x + a.y;
        acc = wmma_f32(a, bm, acc);
    }
    asump += __shfl_xor(asump, 16, 32);   // lanes r and r+16 hold asum[k0+r]

    const int d = d0 + r;
    const float sc = bn1_g[d] * rsqrtf(bn1_v[d] + EPSV);
    const float mn = bn1_m[d];
    const float bb = bn1_b[d];

    float zpart = 0.0f;
#pragma unroll
    for (int j = 0; j < 8; ++j) {
        const int k = k0 + j + 8 * half;
        const float asum_k = __shfl(asump, j + 8 * half, 32);
        const float aggv = acc[j] - asum_k * centers[(size_t)k * D_CH + d];
        zpart += fmaxf((aggv - mn) * sc + bb, 0.0f);
    }
    zpart += __shfl_xor(zpart, 16, 32);   // 16 k-rows of this tile, column d
    if (half == 0) atomicAdd(&z[(size_t)b * D_CH + d], zpart);
}

// ---------------------------------------------------------------------------
// K4: head g = sigmoid((z/32) @ head_w^T + head_b). grid (B), block D
// ---------------------------------------------------------------------------
__global__ void head_kernel(const float* __restrict__ z,
                            const float* __restrict__ head_w,
                            const float* __restrict__ head_b,
                            float* __restrict__ g) {
    const int b = blockIdx.x;
    const int o = threadIdx.x;
    float acc = head_b[o];
    for (int d = 0; d < D_CH; ++d)
        acc += (z[(size_t)b * D_CH + d] * (1.0f / 32.0f)) * head_w[(size_t)o * D_CH + d];
    g[(size_t)b * D_CH + o] = 1.0f / (1.0f + __expf(-acc));
}

// ---------------------------------------------------------------------------
// K5: out = relu(x * (1 + g[b][c])), float4 vectorized
// ---------------------------------------------------------------------------
__global__ void gate_kernel(const float* __restrict__ x,
                            const float* __restrict__ g,
                            float* __restrict__ out) {
    const size_t i4 = (size_t)blockIdx.x * blockDim.x + threadIdx.x;
    const size_t total4 = (size_t)BATCH * D_CH * NPIX / 4;
    if (i4 >= total4) return;
    const size_t bc = (i4 * 4) / NPIX;       // = b*D + c (NPIX % 4 == 0)
    const float gv = 1.0f + g[bc];
    const float4 xv = ((const float4*)x)[i4];
    float4 ov;
    ov.x = fmaxf(xv.x * gv, 0.0f);
    ov.y = fmaxf(xv.y * gv, 0.0f);
    ov.z = fmaxf(xv.z * gv, 0.0f);
    ov.w = fmaxf(xv.w * gv, 0.0f);
    ((float4*)out)[i4] = ov;
}

// ---------------------------------------------------------------------------
extern "C" void kernel_launch(void* const* d_in, const int* in_sizes, int n_in,
                              void* d_out, int out_size, void* d_ws, size_t ws_size,
                              hipStream_t stream) {
    const float* x       = (const float*)d_in[0];
    const float* conv_w  = (const float*)d_in[1];
    const float* bn2_g   = (const float*)d_in[2];
    const float* bn2_b   = (const float*)d_in[3];
    const float* bn2_m   = (const float*)d_in[4];
    const float* bn2_v   = (const float*)d_in[5];
    const float* centers = (const float*)d_in[6];
    const float* scales  = (const float*)d_in[7];
    const float* bn1_g   = (const float*)d_in[8];
    const float* bn1_b   = (const float*)d_in[9];
    const float* bn1_m   = (const float*)d_in[10];
    const float* bn1_v   = (const float*)d_in[11];
    const float* head_w  = (const float*)d_in[12];
    const float* head_b  = (const float*)d_in[13];
    float* out = (float*)d_out;

    // workspace layout
    char* ws = (char*)d_ws;
    float* s_buf = (float*)(ws);                                        // 33.55 MB
    float* a_buf = (float*)(ws + (size_t)BATCH * D_CH * NPIX * 4);      //  4.19 MB
    float* z_buf = a_buf + (size_t)BATCH * KCENT * NPIX;                //  8 KB
    float* g_buf = z_buf + (size_t)BATCH * D_CH;                        //  8 KB

    // K0: zero z accumulator
    init_zero<<<(BATCH * D_CH + 255) / 256, 256, 0, stream>>>(z_buf, BATCH * D_CH);

    // K1: stem GEMM + BN2 + ReLU (LDS-blocked, 64x64 tiles)
    stem_gemm<<<dim3(NPIX / TN, D_CH / TM, BATCH), 128, 0, stream>>>(
        x, conv_w, bn2_g, bn2_b, bn2_m, bn2_v, s_buf);

    // K2: distances + softmax assignment
    assign_softmax<<<dim3(NPIX / 16, BATCH), 32, 0, stream>>>(
        s_buf, centers, scales, a_buf);

    // K3: aggregation GEMM + BN1 + ReLU + K-pool (atomic)
    agg_bn_pool<<<dim3(D_CH / 16, KCENT / 16, BATCH), 32, 0, stream>>>(
        a_buf, s_buf, centers, bn1_g, bn1_b, bn1_m, bn1_v, z_buf);

    // K4: head linear + sigmoid
    head_kernel<<<dim3(BATCH), D_CH, 0, stream>>>(z_buf, head_w, head_b, g_buf);

    // K5: gated residual output
    const size_t total4 = (size_t)BATCH * D_CH * NPIX / 4;
    gate_kernel<<<(unsigned)((total4 + 255) / 256), 256, 0, stream>>>(x, g_buf, out);
}